// FPBlock_7842610283220
// MI455X (gfx1250) — compile-verified
//
#include <hip/hip_runtime.h>

typedef __attribute__((ext_vector_type(2))) float v2f;
typedef __attribute__((ext_vector_type(8))) float v8f;

#define ALPHA 0.3f
#define EPSD  1e-10f

constexpr int B   = 4;
constexpr int N1  = 16384;          // query points   (inputs2 rows)
constexpr int N2  = 2048;           // source points  (inputs1 rows)
constexpr int C1  = 128;            // feat2 channels
constexpr int C2  = 256;            // feat1 channels
constexpr int F1  = 256;
constexpr int F2  = 256;
constexpr int IN1_STRIDE = 3 + C2;  // 259
constexpr int IN2_STRIDE = 3 + C1;  // 131
constexpr int OUT_STRIDE = 3 + F2;  // 259
constexpr int CIN = C1 + C2;        // 384 (GEMM1 K)
constexpr int MT  = 64;             // query rows per workgroup
constexpr int A_STRIDE  = CIN + 4;  // 388: 388%64==4 -> conflict-free column reads
constexpr int H_STRIDE  = F1 + 4;   // 260: same trick
constexpr int WP_STRIDE = 544;      // interleaved K-pair row stride (2*256+32):
                                    // 544%64==32 -> lane groups 0-15 / 16-31 disjoint banks

__device__ __forceinline__ unsigned umin_(unsigned a, unsigned b){ return a < b ? a : b; }
__device__ __forceinline__ unsigned umax_(unsigned a, unsigned b){ return a > b ? a : b; }

// branchless insert of key into sorted (k0<=k1<=k2) smallest-3: 5 v_min/max_u32
__device__ __forceinline__ void ins3(unsigned& k0, unsigned& k1, unsigned& k2, unsigned key){
  unsigned lo  = umin_(key, k0), hi  = umax_(key, k0); k0 = lo;
  unsigned lo1 = umin_(hi,  k1), hi1 = umax_(hi,  k1); k1 = lo1;
  k2 = umin_(k2, hi1);
}

__device__ __forceinline__ float leaky(float x){ return x >= 0.f ? x : ALPHA * x; }

// Chunked GEMM: acc(16x128 per wave) += A(64 x NCHUNK*16, LDS) @ Wg(NCHUNK*16 x 256, global).
// W is staged in LDS as K-pair interleaved rows: wbuf[p][2n] = W[2p][n], wbuf[p][2n+1] = W[2p+1][n],
// so each B-fragment is ONE contiguous ds_load_b64 into an even-aligned VGPR pair (no repacking).
template<int NCHUNK>
__device__ __forceinline__ void gemm_chunked(
    const float* __restrict__ Abase, int aStride,
    const float* __restrict__ Wg,        // global, row-major, 256 cols
    float* __restrict__ wbuf,            // LDS: 2 x 8 x WP_STRIDE floats
    v8f acc[8],
    int tid, int rowTile, int colHalf, int laneM, int laneK)
{
  auto load_chunk = [&](int ci) {        // 16 rows x 256 cols -> 8 interleaved pair-rows
    const float* src = Wg + ci * 16 * 256;
    float* dst = wbuf + (ci & 1) * (8 * WP_STRIDE);
    #pragma unroll
    for (int t = 0; t < 4; ++t) {
      int i = tid + t * 256;             // output float4 index 0..1023
      int p = i >> 7;                    // pair-row 0..7 (128 float4 per pair-row)
      int j = i & 127;                   // float4 within pair-row
      int n = j << 1;                    // covers cols n, n+1
      const float* r0 = src + (2*p) * 256 + n;   // row 2p
      const float* r1 = r0 + 256;                // row 2p+1
      float4 v = make_float4(r0[0], r1[0], r0[1], r1[1]);
      *(float4*)(dst + p * WP_STRIDE + (j << 2)) = v;
    }
  };

  load_chunk(0);
  const float* Ar = Abase + (rowTile * 16 + laneM) * aStride;
  for (int c = 0; c < NCHUNK; ++c) {
    __syncthreads();                               // chunk c visible; buf (c+1)&1 free
    if (c + 1 < NCHUNK) load_chunk(c + 1);
    const float* Wl = wbuf + (c & 1) * (8 * WP_STRIDE) + ((colHalf * 128 + laneM) << 1);
    const float* Ac = Ar + c * 16;
    #pragma unroll
    for (int k = 0; k < 16; k += 4) {
      v2f a; a.x = Ac[k + laneK]; a.y = Ac[k + laneK + 1];
      const float* Wk = Wl + ((k + laneK) >> 1) * WP_STRIDE;  // k+laneK is even
      #pragma unroll
      for (int nt = 0; nt < 8; ++nt) {
        v2f bb = *(const v2f*)(Wk + (nt << 5));    // rows k+laneK, k+laneK+1 @ col n0+nt*16
        acc[nt] = __builtin_amdgcn_wmma_f32_16x16x4_f32(
            false, a, false, bb, (short)0, acc[nt], false, false);
      }
    }
  }
  __syncthreads();   // all reads of A / wbuf done before caller reuses LDS
}

__global__ __launch_bounds__(256)
void fp_block_fused(const float* __restrict__ in1, const float* __restrict__ in2,
                    const float* __restrict__ W1,  const float* __restrict__ W2,
                    float* __restrict__ out)
{
  // LDS, time-multiplexed (~136 KB total, WGP budget is 320 KB):
  //   ldsA: [phase0-1] cand (2048 float4 = 32KB) + merge scratch (3KB above cand)
  //         [phase2-3] A tile 64 x 388 fp32
  //         [phase4-5] H tile 64 x 260 fp32 (aliases A after GEMM1 completes)
  //   ldsWbuf: double-buffered interleaved W chunk for both GEMMs
  __shared__ __align__(16) float ldsA[MT * A_STRIDE];
  __shared__ __align__(16) float ldsWbuf[2 * 8 * WP_STRIDE];
  __shared__ int   ldsIdx[MT * 4];
  __shared__ float ldsW  [MT * 4];

  const int tid  = threadIdx.x;
  const int b    = blockIdx.y;
  const int row0 = blockIdx.x * MT;

  const float* in1b = in1 + (size_t)b * N2 * IN1_STRIDE;
  const float* in2b = in2 + (size_t)b * N1 * IN2_STRIDE;
  float*       outb = out + (size_t)b * N1 * OUT_STRIDE;

  // ---- phase 0: cache candidate xyz + |c|^2 in LDS -------------------------
  float4* cand = (float4*)ldsA;
  for (int m = tid; m < N2; m += 256) {
    const float* p = in1b + m * IN1_STRIDE;
    float x = p[0], y = p[1], z = p[2];
    cand[m] = make_float4(x, y, z, x*x + y*y + z*z);
  }
  __syncthreads();

  // ---- phase 1a: KNN scan, 4 threads per query, packed key = distbits|idx --
  unsigned* merge = (unsigned*)(ldsA + N2 * 4);   // 256*3 uints, above cand
  {
    const int q    = tid >> 2;                    // 0..63
    const int part = tid & 3;
    const float* p = in2b + (row0 + q) * IN2_STRIDE;
    float qx = p[0], qy = p[1], qz = p[2];
    float qn  = qx*qx + qy*qy + qz*qz;
    float m2x = -2.f*qx, m2y = -2.f*qy, m2z = -2.f*qz;
    unsigned k0 = 0xFFFFFFFFu, k1 = 0xFFFFFFFFu, k2 = 0xFFFFFFFFu;
    const int m0 = part * (N2/4);
    for (int m = m0; m < m0 + N2/4; ++m) {
      float4 c = cand[m];
      float d = qn + c.w;
      d = fmaf(c.x, m2x, d);
      d = fmaf(c.y, m2y, d);
      d = fmaf(c.z, m2z, d);
      d = fmaxf(d, 0.f);                            // keep uint-orderable
      unsigned key = (__float_as_uint(d) & 0xFFFFF800u) | (unsigned)m;  // 11 idx bits
      ins3(k0, k1, k2, key);
    }
    unsigned* mp = merge + tid * 3;
    mp[0] = k0; mp[1] = k1; mp[2] = k2;
  }
  __syncthreads();

  // ---- phase 1b: merge partials, exact dists -> normalized weights ---------
  if (tid < MT) {
    const float* p = in2b + (row0 + tid) * IN2_STRIDE;
    float qx = p[0], qy = p[1], qz = p[2];
    float qn = qx*qx + qy*qy + qz*qz;
    unsigned k0 = ~0u, k1 = ~0u, k2 = ~0u;
    const unsigned* mp = merge + tid * 12;
    #pragma unroll
    for (int j = 0; j < 12; ++j) ins3(k0, k1, k2, mp[j]);
    int idx[3] = { (int)(k0 & 0x7FFu), (int)(k1 & 0x7FFu), (int)(k2 & 0x7FFu) };
    float w[3], s = 0.f;
    #pragma unroll
    for (int k = 0; k < 3; ++k) {
      float4 c = cand[idx[k]];
      float d = qn + c.w - 2.f*(qx*c.x + qy*c.y + qz*c.z);
      d = fmaxf(d, EPSD);
      w[k] = 1.f / d; s += w[k];
    }
    float inv = 1.f / s;
    #pragma unroll
    for (int k = 0; k < 3; ++k) { ldsIdx[tid*4+k] = idx[k]; ldsW[tid*4+k] = w[k]*inv; }
    float* o = outb + (row0 + tid) * OUT_STRIDE;    // pass through xyz2
    o[0] = qx; o[1] = qy; o[2] = qz;
  }
  __syncthreads();

  // ---- phase 2: A = [feat2 | interp]  (overwrites cand; cand is dead) ------
  for (int i = tid; i < MT * C1; i += 256) {
    int r = i >> 7, c = i & (C1 - 1);
    ldsA[r * A_STRIDE + c] = in2b[(row0 + r) * IN2_STRIDE + 3 + c];
  }
  {
    const int col = tid;                            // 0..255
    const float* fb = in1b + 3 + col;               // feat1 gathers: L2-resident
    for (int j = 0; j < MT; ++j) {
      int   i0 = ldsIdx[j*4+0], i1 = ldsIdx[j*4+1], i2 = ldsIdx[j*4+2];
      float w0 = ldsW[j*4+0],  w1 = ldsW[j*4+1],  w2 = ldsW[j*4+2];
      float v = w0 * fb[i0 * IN1_STRIDE]
              + w1 * fb[i1 * IN1_STRIDE]
              + w2 * fb[i2 * IN1_STRIDE];
      ldsA[j * A_STRIDE + C1 + col] = v;
    }
  }
  __syncthreads();

  // ---- WMMA tiling: 8 waves = 4 row-tiles x 2 col-halves -------------------
  const int wave    = tid >> 5;
  const int lane    = tid & 31;
  const int rowTile = wave & 3;          // 16-row tile within the 64-row block
  const int colHalf = wave >> 2;         // which 128 of 256 output cols
  const int laneM   = lane & 15;         // ISA: A lanes 0-15 / 16-31 both M=0..15
  const int laneK   = (lane >> 4) << 1;  // ISA: VGPR0 holds K={0,2}, VGPR1 K={1,3}

  // ---- phase 3: GEMM1  acc = A(64x384) @ W1(384x256) -----------------------
  v8f acc[8] = {};
  gemm_chunked<CIN/16>(ldsA, A_STRIDE, W1, ldsWbuf, acc,
                       tid, rowTile, colHalf, laneM, laneK);

  // ---- phase 4: H = leaky(acc) into LDS (aliases A region) -----------------
  float* H = ldsA;
  {
    const int rbase = rowTile*16 + ((lane >> 4) << 3);  // D: VGPR r -> M=r (+8 hi lanes)
    const int nbase = colHalf*128 + laneM;
    #pragma unroll
    for (int nt = 0; nt < 8; ++nt)
      #pragma unroll
      for (int r = 0; r < 8; ++r)
        H[(rbase + r) * H_STRIDE + nbase + nt*16] = leaky(acc[nt][r]);
  }
  __syncthreads();

  // ---- phase 5: GEMM2  acc2 = H(64x256) @ W2(256x256) ----------------------
  v8f acc2[8] = {};
  gemm_chunked<F1/16>(H, H_STRIDE, W2, ldsWbuf, acc2,
                      tid, rowTile, colHalf, laneM, laneK);

  // ---- phase 6: out[..., 3:] = leaky(acc2) ---------------------------------
  {
    const int rbase = row0 + rowTile*16 + ((lane >> 4) << 3);
    const int nbase = colHalf*128 + laneM;
    #pragma unroll
    for (int nt = 0; nt < 8; ++nt)
      #pragma unroll
      for (int r = 0; r < 8; ++r)
        outb[(rbase + r) * OUT_STRIDE + 3 + nbase + nt*16] = leaky(acc2[nt][r]);
  }
}

extern "C" void kernel_launch(void* const* d_in, const int* in_sizes, int n_in,
                              void* d_out, int out_size, void* d_ws, size_t ws_size,
                              hipStream_t stream) {
  const float* in1 = (const float*)d_in[0];   // (B, N2, 3+C2)
  const float* in2 = (const float*)d_in[1];   // (B, N1, 3+C1)
  const float* W1  = (const float*)d_in[2];   // (384, 256)
  const float* W2  = (const float*)d_in[3];   // (256, 256)
  float* out = (float*)d_out;                 // (B, N1, 259)
  (void)in_sizes; (void)n_in; (void)out_size; (void)d_ws; (void)ws_size;

  dim3 grid(N1 / MT, B);   // 256 x 4 workgroups
  dim3 block(256);         // 8 wave32 waves
  fp_block_fused<<<grid, block, 0, stream>>>(in1, in2, W1, W2, out);
}